// Predictor_80539226735106
// MI455X (gfx1250) — compile-verified
//
#include <hip/hip_runtime.h>

// ---------------------------------------------------------------------------
// Edge-score predictor:  out[e] = x[src[e]]@W_src + x[dst[e]]@W_dst + b
// Restructured as:
//   (A) Y[n, 0:64]   = X[n] @ W[0:256]      (node-level GEMM, WMMA f32)
//       Y[n, 64:128] = X[n] @ W[256:512]
//   (B) out[e] = Y[src[e], 0:64] + Y[dst[e], 64:128] + b   (L2-hot gather)
//
// Roofline: GEMM = 6.55 GFLOP fp32; HBM traffic ~0.23 GB total (X read once,
// Y = 51.2 MB lives in the 192 MB L2 for the gather phase).
// ---------------------------------------------------------------------------

typedef __attribute__((ext_vector_type(2))) float v2f;
typedef __attribute__((ext_vector_type(4))) float v4f;
typedef __attribute__((ext_vector_type(8))) float v8f;

#define IN_DIM   256
#define NCLS     64
#define NCAT     128          // 2 * NCLS (concat of src/dst projections)
#define LDS_STR  260          // 256 + 4 pad -> A-frag ds_load_b64 is conflict-free

// ---------------------------------------------------------------------------
// Kernel A: Y[M x 128] = X[M x 256] @ Wcat[256 x 128]
//   Wcat[:, 0:64] = W[0:256, :],  Wcat[:, 64:128] = W[256:512, :]
// One block (8 waves) per 16-node tile; wave w owns N-tile w (16 cols).
// 64 chained v_wmma_f32_16x16x4_f32 per wave (K = 256).
// ---------------------------------------------------------------------------
__global__ __launch_bounds__(256) void gnn_gemm_kernel(
    const float* __restrict__ X,   // [nNodes][256]
    const float* __restrict__ W,   // [512][64] row-major
    float* __restrict__ Y,         // [nNodes][128]
    int nNodes)
{
    __shared__ float ldsA[16 * LDS_STR];

    const int t     = threadIdx.x;        // 0..255
    const int mBase = blockIdx.x * 16;

    // --- Stage 16x256 X tile into LDS (coalesced float4 loads) -------------
    #pragma unroll
    for (int i = 0; i < 4; ++i) {
        int f   = i * 256 + t;            // float4 unit index 0..1023
        int row = f >> 6;                 // 0..15
        int c4  = (f & 63) << 2;          // 0..252 step 4
        int gr  = mBase + row;
        if (gr >= nNodes) gr = nNodes - 1;            // clamp tail reads
        v4f v = *(const v4f*)(X + (size_t)gr * IN_DIM + c4);
        *(v4f*)(&ldsA[row * LDS_STR + c4]) = v;
    }
    __syncthreads();

    const int lane = t & 31;
    const int wave = t >> 5;              // N-tile 0..7
    const int half = lane >> 4;           // 0/1 (K-pair selector per ISA layout)
    const int mrow = lane & 15;           // M (for A) / N (for B,C,D) index
    const int n    = wave * 16 + mrow;    // global output column 0..127

    // Column pointer into W for this lane (wave-uniform branch):
    //   n <  64 : W_src column n       -> W[k*64 + n]
    //   n >= 64 : W_dst column n - 64  -> W[(256+k)*64 + (n-64)]
    const float* wcol = (n < NCLS) ? (W + n)
                                   : (W + (size_t)IN_DIM * NCLS + (n - NCLS));

    const float* aRow = &ldsA[mrow * LDS_STR];   // lane's A row in LDS

    v8f acc = {};

#if defined(__HIP_DEVICE_COMPILE__) && __has_builtin(__builtin_amdgcn_wmma_f32_16x16x4_f32)
    // ---- WMMA f32 path: 64 chained v_wmma_f32_16x16x4_f32 ----------------
    #pragma unroll 8
    for (int kb = 0; kb < IN_DIM; kb += 4) {
        const int k = kb + 2 * half;
        // A frag: A[m = lane%16][k, k+1]  (one ds_load_b64, bank-conflict-free)
        v2f a = *(const v2f*)(aRow + k);
        // B frag: B[k, k+1][n]  (two b32 loads from cache-hot W)
        v2f b;
        b.x = wcol[(size_t)k * NCLS];
        b.y = wcol[(size_t)(k + 1) * NCLS];
        acc = __builtin_amdgcn_wmma_f32_16x16x4_f32(
                  /*neg_a=*/false, a, /*neg_b=*/false, b,
                  /*c_mod=*/(short)0, acc, /*reuse_a=*/false, /*reuse_b=*/false);
    }
#else
    // ---- VALU fallback (also parsed by the host pass) ---------------------
    (void)aRow;
    #pragma unroll 4
    for (int k = 0; k < IN_DIM; ++k) {
        const float wv = wcol[(size_t)k * NCLS];
        #pragma unroll
        for (int r = 0; r < 8; ++r)
            acc[r] = fmaf(ldsA[(8 * half + r) * LDS_STR + k], wv, acc[r]);
    }
#endif

    // --- Store D: VGPR r, lane L -> Y[mBase + r + 8*half][n] ---------------
    const int mOff = mBase + 8 * half;
    float* yBase = Y + (size_t)mOff * NCAT + n;
    if (mBase + 16 <= nNodes) {
        // Full tile (always the case when nNodes % 16 == 0): branch-free.
        #pragma unroll
        for (int r = 0; r < 8; ++r)
            yBase[(size_t)r * NCAT] = acc[r];
    } else {
        // Tail tile: per-row guard.
        #pragma unroll
        for (int r = 0; r < 8; ++r)
            if (mOff + r < nNodes)
                yBase[(size_t)r * NCAT] = acc[r];
    }
}

// ---------------------------------------------------------------------------
// Kernel B: out[e][c] = Y[src[e]][c] + Y[dst[e]][64 + c] + b[c]
// 16 threads per edge, float4 per thread. Y (51.2 MB) is L2-resident.
// ---------------------------------------------------------------------------
__global__ __launch_bounds__(256) void gnn_edge_kernel(
    const float* __restrict__ Y,
    const int*   __restrict__ src,
    const int*   __restrict__ dst,
    const float* __restrict__ bias,
    float*       __restrict__ out,
    int nEdges)
{
    const long long idx = (long long)blockIdx.x * 256 + threadIdx.x;
    if (idx >= (long long)nEdges * 16) return;

    const int e  = (int)(idx >> 4);
    const int c4 = (int)(idx & 15) << 2;      // 0..60 step 4

    const int s = src[e];
    const int d = dst[e];

    v4f ys = *(const v4f*)(Y + (size_t)s * NCAT + c4);
    v4f yd = *(const v4f*)(Y + (size_t)d * NCAT + NCLS + c4);
    v4f bb = *(const v4f*)(bias + c4);

    v4f r = ys + yd + bb;
    *(v4f*)(out + (size_t)e * NCLS + c4) = r;
}

// ---------------------------------------------------------------------------
extern "C" void kernel_launch(void* const* d_in, const int* in_sizes, int n_in,
                              void* d_out, int out_size, void* d_ws, size_t ws_size,
                              hipStream_t stream) {
    const float* X    = (const float*)d_in[0];   // [nNodes*256]
    const int*   src  = (const int*)  d_in[1];   // [nEdges]
    const int*   dst  = (const int*)  d_in[2];   // [nEdges]
    const float* W    = (const float*)d_in[3];   // [512*64]
    const float* bias = (const float*)d_in[4];   // [64]
    float*       out  = (float*)d_out;           // [nEdges*64]
    float*       Y    = (float*)d_ws;            // needs nNodes*128*4 = 51.2 MB

    const int nNodes = in_sizes[0] / IN_DIM;     // 100000
    const int nEdges = in_sizes[1];              // 300000

    const int mTiles = (nNodes + 15) / 16;
    gnn_gemm_kernel<<<mTiles, 256, 0, stream>>>(X, W, Y, nNodes);

    const long long tot    = (long long)nEdges * 16;
    const int       blocks = (int)((tot + 255) / 256);
    gnn_edge_kernel<<<blocks, 256, 0, stream>>>(Y, src, dst, bias, out, nEdges);
}